// EnhancedUltra_74251394613542
// MI455X (gfx1250) — compile-verified
//
#include <hip/hip_runtime.h>
#include <cstdint>
#include <cstddef>

// Problem constants (match reference file).
#define BQ   4096
#define RR   500
#define DD   64
#define NN   100000
#define EE   3200000
#define FDIM 132          // 2*D + 4

typedef float v2f __attribute__((ext_vector_type(2)));
typedef float v8f __attribute__((ext_vector_type(8)));

// ---------------------------------------------------------------------------
// Workspace init: node2rep = INT_MAX, cnt = 0, rel_freq = 0
// ---------------------------------------------------------------------------
__global__ void k_init(int* __restrict__ node2rep, float* __restrict__ cnt,
                       float* __restrict__ rel_freq) {
  int i = blockIdx.x * blockDim.x + threadIdx.x;
  int stride = gridDim.x * blockDim.x;
  for (int j = i; j < NN; j += stride) node2rep[j] = 0x7fffffff;
  for (int j = i; j < BQ * RR; j += stride) cnt[j] = 0.0f;
  for (int j = i; j < RR; j += stride) rel_freq[j] = 0.0f;
}

// ---------------------------------------------------------------------------
// Map each queried entity to the smallest batch index that queries it.
// ---------------------------------------------------------------------------
__global__ void k_map(const int* __restrict__ qents, int* __restrict__ node2rep) {
  int b = blockIdx.x * blockDim.x + threadIdx.x;
  if (b < BQ) atomicMin(&node2rep[qents[b]], b);
}

// ---------------------------------------------------------------------------
// Edge pass: per-type counts at tracked nodes + global relation frequency.
// rel_freq privatized in LDS (500 counters/block). All adds integer-valued
// floats < 2^24 -> exact, deterministic regardless of atomic order.
// ---------------------------------------------------------------------------
__global__ void k_edges(const int* __restrict__ edge_index,
                        const int* __restrict__ edge_type,
                        const int* __restrict__ node2rep,
                        float* __restrict__ cnt,
                        float* __restrict__ rel_freq) {
  __shared__ float relf[RR];
  for (int i = threadIdx.x; i < RR; i += blockDim.x) relf[i] = 0.0f;
  __syncthreads();

  int stride = gridDim.x * blockDim.x;
  for (int e = blockIdx.x * blockDim.x + threadIdx.x; e < EE; e += stride) {
    int r = edge_type[e];
    atomicAdd(&relf[r], 1.0f);
    int s = edge_index[e];        // src
    int d = edge_index[EE + e];   // dst
    int rs = node2rep[s];
    if (rs < BQ) atomicAdd(&cnt[(size_t)rs * RR + r], 1.0f);
    if (d != s) {                 // self-loop contributes only once (src side)
      int rd = node2rep[d];
      if (rd < BQ) atomicAdd(&cnt[(size_t)rd * RR + r], 1.0f);
    }
  }
  __syncthreads();
  for (int i = threadIdx.x; i < RR; i += blockDim.x) {
    float v = relf[i];
    if (v != 0.0f) atomicAdd(&rel_freq[i], v);
  }
}

// ---------------------------------------------------------------------------
// Feature build: one block per batch row. Streams relation_embeddings[b]
// (128 KB) once with float4 coalesced loads -> this is the 524 MB / 23.3 TB/s
// roofline kernel. ent_emb = (cnt . emb) / max(tot,1); deterministic fixed
// slicing.
// ---------------------------------------------------------------------------
__global__ void __launch_bounds__(256)
k_feats(const float* __restrict__ emb, const int* __restrict__ qrels,
        const int* __restrict__ qents, const int* __restrict__ node2rep,
        const float* __restrict__ cnt, const float* __restrict__ rel_freq,
        float* __restrict__ feats) {
  __shared__ float cl[RR];
  __shared__ float partial[16 * 64];
  __shared__ float tot_s;

  int b = blockIdx.x;
  int tid = threadIdx.x;
  if (tid == 0) tot_s = 0.0f;
  __syncthreads();

  int rep = node2rep[qents[b]];   // always < BQ (b itself wrote one)
  float lt = 0.0f;
  for (int i = tid; i < RR; i += 256) {
    float c = cnt[(size_t)rep * RR + i];
    cl[i] = c;
    lt += c;
  }
  atomicAdd(&tot_s, lt);          // integer-valued -> exact
  __syncthreads();
  float tot = tot_s;

  // 16 slices x 16 column-quads; float4 loads, rows interleaved by 16.
  int q = tid & 15;               // column quad: columns q*4 .. q*4+3
  int slice = tid >> 4;           // 0..15
  const float4* embv = (const float4*)(emb + (size_t)b * RR * DD);
  float4 acc = make_float4(0.f, 0.f, 0.f, 0.f);
  for (int r = slice; r < RR; r += 16) {
    float c = cl[r];
    float4 v = embv[r * 16 + q];
    acc.x += c * v.x; acc.y += c * v.y; acc.z += c * v.z; acc.w += c * v.w;
  }
  partial[slice * 64 + q * 4 + 0] = acc.x;
  partial[slice * 64 + q * 4 + 1] = acc.y;
  partial[slice * 64 + q * 4 + 2] = acc.z;
  partial[slice * 64 + q * 4 + 3] = acc.w;
  __syncthreads();

  if (tid < 64) {
    float s = 0.0f;
    for (int sl = 0; sl < 16; ++sl) s += partial[sl * 64 + tid];
    float ent = (tot > 0.0f) ? (s / fmaxf(tot, 1.0f)) : 0.0f;
    int myqr = qrels[b];
    float rel = emb[((size_t)b * RR + myqr) * DD + tid];
    float* f = feats + (size_t)b * FDIM;
    f[tid] = rel;          // rel_emb
    f[64 + tid] = ent;     // ent_emb
    if (tid == 0) {
      float rfn = fminf(rel_freq[myqr] * (1.0f / (float)EE), 1.0f);
      float edn = fminf(tot * (1.0f / (float)EE), 1.0f);
      float dens = fminf((float)EE / ((float)NN * (float)NN), 1.0f);
      f[128] = rfn; f[129] = edn; f[130] = rfn; f[131] = dens;
    }
  }
}

// ---------------------------------------------------------------------------
// MLP via V_WMMA_F32_16X16X4_F32. One wave (32 threads) per 16 batch rows.
// Operand layouts per CDNA5 ISA 7.12.2:
//   A 16x4 : lanes 0-15 -> M=lane, K=0(x)/1(y); lanes 16-31 -> M=lane-16, K=2/3
//   B 4x16 : VGPR0: K=0 (lanes 0-15) / K=2 (16-31); VGPR1: K=1 / K=3; N=lane%16
//   D 16x16: VGPR r: M=r (lanes 0-15) / M=r+8 (16-31); N=lane%16
// ---------------------------------------------------------------------------
__global__ void __launch_bounds__(32)
k_mlp(const float* __restrict__ feats,
      const float* __restrict__ W1, const float* __restrict__ b1,
      const float* __restrict__ W2, const float* __restrict__ b2,
      const float* __restrict__ W3, const float* __restrict__ b3,
      const float* __restrict__ W4, const float* __restrict__ b4,
      float* __restrict__ out) {
  __shared__ float ft[16 * FDIM];   // stride 132 (bank = 4*row + k, conflict-free)
  __shared__ float h1[16 * 65];
  __shared__ float h2[16 * 33];
  __shared__ float gg[16 * 17];

  int lane = threadIdx.x;
  int half = lane >> 4;
  int ln = lane & 15;
  int row0 = blockIdx.x * 16;

  for (int i = lane; i < 16 * FDIM; i += 32) {
    int r = i / FDIM, c = i % FDIM;
    ft[i] = feats[(size_t)(row0 + r) * FDIM + c];
  }
  __syncthreads();

  // ---- Layer 1: [16 x 132] @ [132 x 64] -> relu -> h1 ----
  for (int nt = 0; nt < 4; ++nt) {
    v8f acc = {};
    int col = nt * 16 + ln;
    for (int kc = 0; kc < 33; ++kc) {
      int kb = kc * 4 + half * 2;
      v2f a, bb;
      a.x = ft[ln * FDIM + kb];
      a.y = ft[ln * FDIM + kb + 1];
      bb.x = W1[kb * 64 + col];
      bb.y = W1[(kb + 1) * 64 + col];
      acc = __builtin_amdgcn_wmma_f32_16x16x4_f32(false, a, false, bb,
                                                  (short)0, acc, false, false);
    }
    float bias = b1[col];
    for (int r = 0; r < 8; ++r) {
      float v = acc[r] + bias;
      h1[(r + half * 8) * 65 + col] = v > 0.0f ? v : 0.0f;
    }
  }
  __syncthreads();

  // ---- Layer 2: [16 x 64] @ [64 x 32] -> relu -> h2 ----
  for (int nt = 0; nt < 2; ++nt) {
    v8f acc = {};
    int col = nt * 16 + ln;
    for (int kc = 0; kc < 16; ++kc) {
      int kb = kc * 4 + half * 2;
      v2f a, bb;
      a.x = h1[ln * 65 + kb];
      a.y = h1[ln * 65 + kb + 1];
      bb.x = W2[kb * 32 + col];
      bb.y = W2[(kb + 1) * 32 + col];
      acc = __builtin_amdgcn_wmma_f32_16x16x4_f32(false, a, false, bb,
                                                  (short)0, acc, false, false);
    }
    float bias = b2[col];
    for (int r = 0; r < 8; ++r) {
      float v = acc[r] + bias;
      h2[(r + half * 8) * 33 + col] = v > 0.0f ? v : 0.0f;
    }
  }
  __syncthreads();

  // ---- Layer 3: [16 x 32] @ [32 x 16] -> relu -> gg ----
  {
    v8f acc = {};
    for (int kc = 0; kc < 8; ++kc) {
      int kb = kc * 4 + half * 2;
      v2f a, bb;
      a.x = h2[ln * 33 + kb];
      a.y = h2[ln * 33 + kb + 1];
      bb.x = W3[kb * 16 + ln];
      bb.y = W3[(kb + 1) * 16 + ln];
      acc = __builtin_amdgcn_wmma_f32_16x16x4_f32(false, a, false, bb,
                                                  (short)0, acc, false, false);
    }
    float bias = b3[ln];
    for (int r = 0; r < 8; ++r) {
      float v = acc[r] + bias;
      gg[(r + half * 8) * 17 + ln] = v > 0.0f ? v : 0.0f;
    }
  }
  __syncthreads();

  // ---- Layer 4: [16 x 16] @ [16 x 1] + b4 -> sigmoid ----
  if (lane < 16) {
    float s = b4[0];
    for (int j = 0; j < 16; ++j) s += gg[lane * 17 + j] * W4[j];
    out[row0 + lane] = 1.0f / (1.0f + expf(-s));
  }
}

// ---------------------------------------------------------------------------
static inline size_t align_up(size_t x) { return (x + 255) & ~(size_t)255; }

extern "C" void kernel_launch(void* const* d_in, const int* in_sizes, int n_in,
                              void* d_out, int out_size, void* d_ws, size_t ws_size,
                              hipStream_t stream) {
  const float* emb   = (const float*)d_in[0];   // [B, R, D]
  const int*   qrels = (const int*)d_in[1];     // [B]
  const int*   qents = (const int*)d_in[2];     // [B]
  const int*   eidx  = (const int*)d_in[3];     // [2, E]
  const int*   etype = (const int*)d_in[4];     // [E]
  const float* W1 = (const float*)d_in[5];
  const float* B1 = (const float*)d_in[6];
  const float* W2 = (const float*)d_in[7];
  const float* B2 = (const float*)d_in[8];
  const float* W3 = (const float*)d_in[9];
  const float* B3 = (const float*)d_in[10];
  const float* W4 = (const float*)d_in[11];
  const float* B4 = (const float*)d_in[12];
  float* out = (float*)d_out;

  char* w = (char*)d_ws;
  size_t off = 0;
  int*   node2rep = (int*)(w + off);   off += align_up((size_t)NN * 4);
  float* cnt      = (float*)(w + off); off += align_up((size_t)BQ * RR * 4);
  float* rel_freq = (float*)(w + off); off += align_up((size_t)RR * 4);
  float* feats    = (float*)(w + off); off += align_up((size_t)BQ * FDIM * 4);
  (void)ws_size; (void)in_sizes; (void)n_in; (void)out_size;

  k_init <<<1024, 256, 0, stream>>>(node2rep, cnt, rel_freq);
  k_map  <<<(BQ + 255) / 256, 256, 0, stream>>>(qents, node2rep);
  k_edges<<<2048, 256, 0, stream>>>(eidx, etype, node2rep, cnt, rel_freq);
  k_feats<<<BQ, 256, 0, stream>>>(emb, qrels, qents, node2rep, cnt, rel_freq, feats);
  k_mlp  <<<BQ / 16, 32, 0, stream>>>(feats, W1, B1, W2, B2, W3, B3, W4, B4, out);
}